// LightGCNConv_7146825581232
// MI455X (gfx1250) — compile-verified
//
#include <hip/hip_runtime.h>

#define DIM 64

typedef float v2f __attribute__((ext_vector_type(2)));
typedef float v8f __attribute__((ext_vector_type(8)));

// ---------------------------------------------------------------------------
// Kernel 1: zero the accumulator (d_out) with wide stores.
// ---------------------------------------------------------------------------
__global__ __launch_bounds__(256) void zero_kernel(float4* __restrict__ out, unsigned n4) {
    unsigned i = blockIdx.x * 256u + threadIdx.x;
    if (i < n4) out[i] = make_float4(0.f, 0.f, 0.f, 0.f);
}

// ---------------------------------------------------------------------------
// Kernel 2: edge scatter. 16 lanes per edge, float4 (b128) gather of h[src],
// scale by edge weight, hardware f32 atomics into agg[dst]. h table and agg
// are both L2-resident (76.8 MB each, 192 MB L2), so atomics resolve in L2.
// ---------------------------------------------------------------------------
__global__ __launch_bounds__(256) void scatter_kernel(
    const float* __restrict__ user_emb,
    const float* __restrict__ item_emb,
    const int*   __restrict__ src,
    const int*   __restrict__ dst,
    const float* __restrict__ ew,
    float*       __restrict__ out,
    int n_users, int n_edges)
{
    unsigned gid  = blockIdx.x * 256u + threadIdx.x;   // up to 20M, fits u32
    int      e    = (int)(gid >> 4);
    int      lane = (int)(gid & 15u);
    if (e >= n_edges) return;

    int   s = src[e];
    int   d = dst[e];
    float w = ew[e];

    const float* row = (s < n_users)
                           ? (user_emb + (size_t)s * DIM)
                           : (item_emb + (size_t)(s - n_users) * DIM);

    float4 v = ((const float4*)row)[lane];             // global_load_b128

    float* o = out + (size_t)d * DIM + lane * 4;
    unsafeAtomicAdd(o + 0, v.x * w);                   // global_atomic_add_f32
    unsafeAtomicAdd(o + 1, v.y * w);
    unsafeAtomicAdd(o + 2, v.z * w);
    unsafeAtomicAdd(o + 3, v.w * w);
}

// ---------------------------------------------------------------------------
// Kernel 3: row L2-normalize, 1 wave32 per 16 nodes.
// Gram trick: acc = A * A^T via 16x V_WMMA_F32_16X16X4_F32 over K-chunks of 4;
// diag(acc) = per-row sum of squares (exact f32). A-fragment layout (ISA
// 7.12.2, 32-bit A 16x4): lane l holds row (l&15), K-pair {0,1} for l<16 and
// {2,3} for l>=16 of each chunk. B mirrors the same layout for 4x16, so the
// identical register pair is fed as both A and B.
// Diagonal: node i<8 -> lane i, acc[i]; node i>=8 -> lane i+16, acc[i-8].
// ---------------------------------------------------------------------------
__global__ __launch_bounds__(32) void normalize_kernel(float* __restrict__ out, int n_nodes)
{
    const int lane = threadIdx.x;                 // 0..31
    const int m    = lane & 15;                   // row within tile
    const int koff = (lane < 16) ? 0 : 2;         // K offset inside a chunk

    int r = blockIdx.x * 16 + m;                  // node index for this lane
    const bool valid = (r < n_nodes);
    if (r >= n_nodes) r = n_nodes - 1;            // clamp: keep EXEC all-ones for WMMA
    float* rowp = out + (size_t)r * DIM;

    // Load A fragments: 16 chunks x float2 per lane (the whole 16x64 tile).
    v2f a[16];
#pragma unroll
    for (int c = 0; c < 16; ++c)
        a[c] = *(const v2f*)(rowp + 4 * c + koff);

    v8f acc = {};
#pragma unroll
    for (int c = 0; c < 16; ++c)
        acc = __builtin_amdgcn_wmma_f32_16x16x4_f32(
            /*neg_a=*/false, a[c], /*neg_b=*/false, a[c],
            /*c_mod=*/(short)0, acc, /*reuse_a=*/false, /*reuse_b=*/false);

    // Extract diagonal -> inverse norms, exchange through 64B of LDS.
    __shared__ float s_inv[16];
    if (lane < 8) {
        float ss = acc[lane];
        s_inv[lane] = 1.0f / fmaxf(sqrtf(ss), 1e-12f);
    } else if (lane >= 24) {
        float ss = acc[lane - 24];
        s_inv[lane - 16] = 1.0f / fmaxf(sqrtf(ss), 1e-12f);
    }
    __syncthreads();
    float inv = s_inv[m];

    // Scale and stream out (non-temporal: result is never re-read).
#pragma unroll
    for (int c = 0; c < 16; ++c) {
        if (valid) {
            v2f sv = a[c] * inv;
            __builtin_nontemporal_store(sv, (v2f*)(rowp + 4 * c + koff));
        }
    }
}

// ---------------------------------------------------------------------------
extern "C" void kernel_launch(void* const* d_in, const int* in_sizes, int n_in,
                              void* d_out, int out_size, void* d_ws, size_t ws_size,
                              hipStream_t stream)
{
    const float* user_emb = (const float*)d_in[0];
    const float* item_emb = (const float*)d_in[1];
    const int*   src      = (const int*)d_in[2];
    const int*   dst      = (const int*)d_in[3];
    const float* ew       = (const float*)d_in[4];
    float*       out      = (float*)d_out;

    const int n_users = in_sizes[0] / DIM;
    const int n_items = in_sizes[1] / DIM;
    const int n_edges = in_sizes[2];
    const int n_nodes = n_users + n_items;

    // 1) zero accumulator
    {
        unsigned n4 = (unsigned)((size_t)n_nodes * DIM / 4);
        unsigned blocks = (n4 + 255u) / 256u;
        zero_kernel<<<blocks, 256, 0, stream>>>((float4*)out, n4);
    }
    // 2) scatter-add messages
    {
        unsigned total  = (unsigned)n_edges * 16u;
        unsigned blocks = (total + 255u) / 256u;
        scatter_kernel<<<blocks, 256, 0, stream>>>(user_emb, item_emb, src, dst, ew,
                                                   out, n_users, n_edges);
    }
    // 3) WMMA-based row normalization, 16 nodes per wave32
    {
        unsigned blocks = (unsigned)((n_nodes + 15) / 16);
        normalize_kernel<<<blocks, 32, 0, stream>>>(out, n_nodes);
    }
}